// MLPGaussianModule_79869211837086
// MI455X (gfx1250) — compile-verified
//
#include <hip/hip_runtime.h>

// MI455X / gfx1250, wave32. Memory-bound kernel (536MB of M @ 23.3TB/s ~ 23us
// floor); fp32 WMMA (v_wmma_f32_16x16x4_f32) keeps compute off the VALU while
// HBM streams. M is loaded nontemporally (read-once), x-scale and the j==t
// diagonal mask are folded in at LDS-staging time so the WMMA inner loop is
// just 2 LDS reads + 1 WMMA per (t, batch-subtile).

typedef __attribute__((ext_vector_type(2))) float v2f;
typedef __attribute__((ext_vector_type(4))) float v4f;
typedef __attribute__((ext_vector_type(8))) float v8f;

#define NVARS 256
#define HID 16
#define BATCH 2048
#define TCH 32                 // t per block (one 128B cacheline of M)
#define BTL 32                 // batch per block
#define JC 4                   // j per pipeline stage
#define NSTAGE (NVARS / JC)    // 64
#define TP 36                  // padded t-stride (floats) per j row (16B-aligned)
#define BSTR (JC * TP + 4)     // 148 floats per b row: odd-ish bank spread, 16B-aligned
#define NEGSLOPE 0.01f

__device__ __forceinline__ float lrelu(float v) {
    return fmaxf(v, 0.0f) + NEGSLOPE * fminf(v, 0.0f);
}

__global__ __launch_bounds__(256) void causal_mlp_fused(
    const float* __restrict__ x,   // (B, NVARS)
    const float* __restrict__ M,   // (B, NVARS, NVARS)  [b, j, t]
    const float* __restrict__ W0,  // (NVARS, HID, NVARS) [t, i, j]
    const float* __restrict__ W1,  // (NVARS, HID, HID)
    const float* __restrict__ W2,  // (NVARS, 1, HID)
    const float* __restrict__ b0,  // (NVARS, HID)
    const float* __restrict__ b1,  // (NVARS, HID)
    const float* __restrict__ b2,  // (NVARS, 1)
    float* __restrict__ out)       // (B, NVARS)
{
    __shared__ float mld[2][BTL * BSTR];    // double-buffered xm stage: 2*4736 f
    __shared__ float hscr[8][HID * 17];     // per-wave 16x16 transpose scratch
    __shared__ float outs[BTL][TCH + 4];    // output staging, padded

    const int tid  = threadIdx.x;
    const int w    = tid >> 5;       // wave id 0..7
    const int lane = tid & 31;
    const int n    = lane & 15;      // N column (batch within subtile)
    const int hi   = lane >> 4;      // half-wave -> K pair select

    const int bbase = blockIdx.x * BTL;
    const int t0    = blockIdx.y * TCH;

    // Layer-0 accumulators: wave owns 4 t-values x 2 batch subtiles.
    v8f acc[4][2];
#pragma unroll
    for (int i = 0; i < 4; ++i) {
        acc[i][0] = v8f{0,0,0,0,0,0,0,0};
        acc[i][1] = v8f{0,0,0,0,0,0,0,0};
    }

    // ---- stage fetch: 4 x 128b per thread; fold x[b,j] and diag mask ----
    auto fetch = [&](int s, v4f* r) {
        const int jbase = s * JC;
#pragma unroll
        for (int q = 0; q < 4; ++q) {
            int f    = q * 256 + tid;
            int t4   = f & 7;          // which float4 along t
            int pair = f >> 3;
            int bl   = pair & 31;
            int jl   = pair >> 5;      // 0..3
            int gb   = bbase + bl;
            int gj   = jbase + jl;
            const v4f* mp =
                (const v4f*)(M + ((size_t)gb * NVARS + gj) * NVARS + t0 + t4 * 4);
            v4f   mv = __builtin_nontemporal_load(mp);   // M is read-once
            float xs = x[gb * NVARS + gj];
            mv *= xs;
            int dt = gj - (t0 + t4 * 4);                 // zero diagonal j==t
            if (dt == 0) mv.x = 0.0f;
            if (dt == 1) mv.y = 0.0f;
            if (dt == 2) mv.z = 0.0f;
            if (dt == 3) mv.w = 0.0f;
            r[q] = mv;
        }
    };
    auto stash = [&](float* buf, const v4f* r) {
#pragma unroll
        for (int q = 0; q < 4; ++q) {
            int f    = q * 256 + tid;
            int t4   = f & 7;
            int pair = f >> 3;
            int bl   = pair & 31;
            int jl   = pair >> 5;
            *(v4f*)(buf + bl * BSTR + jl * TP + t4 * 4) = r[q];
        }
    };

    // ---- layer-0 WMMA over one staged j-chunk ----
    // A (16x4 f32): lane m=n, vgprs hold K = 2*hi, 2*hi+1
    // B (4x16 f32): lane col=n, vgprs hold K = 2*hi, 2*hi+1
    auto compute = [&](const float* buf, int s) {
        const int jbase = s * JC;
        const int k0 = 2 * hi;
#pragma unroll
        for (int ti = 0; ti < 4; ++ti) {
            int tl = w * 4 + ti;
            int t  = t0 + tl;
            const float* ap = W0 + (size_t)t * (HID * NVARS) + n * NVARS + jbase + k0;
            v2f a; a.x = ap[0]; a.y = ap[1];
#pragma unroll
            for (int sub = 0; sub < 2; ++sub) {
                int brow = sub * 16 + n;
                v2f bb;
                bb.x = buf[brow * BSTR + k0 * TP + tl];
                bb.y = buf[brow * BSTR + (k0 + 1) * TP + tl];
                acc[ti][sub] = __builtin_amdgcn_wmma_f32_16x16x4_f32(
                    false, a, false, bb, (short)0, acc[ti][sub], false, false);
            }
        }
    };

    // ---- software-pipelined main loop over j ----
    v4f pre[4];
    fetch(0, pre);
    stash(&mld[0][0], pre);
    __syncthreads();
    int cur = 0;
    for (int s = 0; s < NSTAGE; ++s) {
        const bool more = (s + 1 < NSTAGE);
        if (more) fetch(s + 1, pre);       // loads in flight while computing
        compute(&mld[cur][0], s);
        __syncthreads();                   // everyone done reading buf cur^1
        if (more) stash(&mld[cur ^ 1][0], pre);
        __syncthreads();                   // stage visible
        cur ^= 1;
    }

    // ---- epilogue: bias+lrelu, layer1 (WMMA), layer2 (dot), store ----
    float* hs = &hscr[w][0];
#pragma unroll
    for (int ti = 0; ti < 4; ++ti) {
        int tl = w * 4 + ti;
        int t  = t0 + tl;
#pragma unroll
        for (int sub = 0; sub < 2; ++sub) {
            v8f h = acc[ti][sub];
#pragma unroll
            for (int r = 0; r < 8; ++r)
                h[r] = lrelu(h[r] + b0[t * HID + r + 8 * hi]);

            // transpose tile through LDS: element (m = r + 8*hi, n)
#pragma unroll
            for (int r = 0; r < 8; ++r)
                hs[(r + 8 * hi) * 17 + n] = h[r];
            asm volatile("s_wait_dscnt 0x0" ::: "memory");  // wave-local RAW

            v8f h2 = v8f{0,0,0,0,0,0,0,0};
#pragma unroll
            for (int ks = 0; ks < 4; ++ks) {
                int kk = ks * 4 + 2 * hi;
                const float* a1p = W1 + t * (HID * HID) + n * HID + kk;
                v2f a1; a1.x = a1p[0]; a1.y = a1p[1];
                v2f bb; bb.x = hs[kk * 17 + n]; bb.y = hs[(kk + 1) * 17 + n];
                h2 = __builtin_amdgcn_wmma_f32_16x16x4_f32(
                    false, a1, false, bb, (short)0, h2, false, false);
            }

            float part = 0.0f;
#pragma unroll
            for (int r = 0; r < 8; ++r) {
                float v = lrelu(h2[r] + b1[t * HID + r + 8 * hi]);
                part = fmaf(v, W2[t * HID + r + 8 * hi], part);
            }
            float tot = part + __shfl_xor(part, 16, 32) + b2[t];
            if (hi == 0) outs[sub * 16 + n][tl] = tot;
        }
    }
    __syncthreads();

    // coalesced final store: each thread writes one 128b (32 t contiguous per b)
    {
        int bl = tid >> 3;
        int t4 = tid & 7;
        v4f v = *(const v4f*)(&outs[bl][t4 * 4]);
        *(v4f*)(out + (size_t)(bbase + bl) * NVARS + t0 + t4 * 4) = v;
    }
}

extern "C" void kernel_launch(void* const* d_in, const int* in_sizes, int n_in,
                              void* d_out, int out_size, void* d_ws, size_t ws_size,
                              hipStream_t stream) {
    const float* x  = (const float*)d_in[0];
    const float* M  = (const float*)d_in[1];
    const float* W0 = (const float*)d_in[2];
    const float* W1 = (const float*)d_in[3];
    const float* W2 = (const float*)d_in[4];
    const float* b0 = (const float*)d_in[5];
    const float* b1 = (const float*)d_in[6];
    const float* b2 = (const float*)d_in[7];
    float* out = (float*)d_out;

    dim3 grid(BATCH / BTL, NVARS / TCH, 1);   // 64 x 8
    dim3 block(256, 1, 1);
    causal_mlp_fused<<<grid, block, 0, stream>>>(x, M, W0, W1, W2, b0, b1, b2, out);
}